// EvolveGCN_reg_3719441678531
// MI455X (gfx1250) — compile-verified
//
#include <hip/hip_runtime.h>
#include <math.h>

// ---------------------------------------------------------------------------
// EvolveGCN-style recurrence, MI455X (gfx1250, wave32).
//
// Algebraic reduction:  out[t] = (AH @ W_new) @ lin_w + b
//                              =  AH @ (W_new @ lin_w) + b
//                              =  segment_sum(val * (X_t[col] . w_eff)) + b
// -> dominant loop is 1 float atomicAdd per edge (L2-resident target),
//    HBM-bound on the edge streams (~360 MB total, ~15us floor @ 23.3 TB/s).
// GRU 16x16 matmuls use V_WMMA_F32_16X16X4_F32 (f32 in/out, exact precision).
// ---------------------------------------------------------------------------

typedef float v2f __attribute__((ext_vector_type(2)));
typedef float v8f __attribute__((ext_vector_type(8)));

#define TT 8
#define NN 100000
#define EE 3200000
#define F 16

// ---------------------------------------------------------------------------
// init: W_cur = W_init  (must rerun every launch; ws is not preserved state)
// ---------------------------------------------------------------------------
__global__ __launch_bounds__(256) void init_w_kernel(const float* __restrict__ W_init,
                                                     float* __restrict__ W_cur) {
    int i = threadIdx.x;  // 256 threads == 256 elements
    W_cur[i] = W_init[i];
}

// ---------------------------------------------------------------------------
// y[i] = (X_t[i] . p) / ||p||
// ---------------------------------------------------------------------------
__global__ __launch_bounds__(256) void y_kernel(const float* __restrict__ Xt,
                                                const float* __restrict__ p,
                                                float* __restrict__ y, int n) {
    int i = blockIdx.x * blockDim.x + threadIdx.x;
    if (i >= n) return;
    float nrm = 0.f, d = 0.f;
#pragma unroll
    for (int f = 0; f < F; ++f) {
        float pv = p[f];
        nrm += pv * pv;
        d += Xt[(size_t)i * F + f] * pv;
    }
    y[i] = d / sqrtf(nrm);
}

// ---------------------------------------------------------------------------
// Deterministic iterative top-16 (desc, ties -> lower index == lax.top_k),
// then Xs[f][k] = X_t[idx_k][f] * y_k. Single block; y is scratch (destroyed).
// ---------------------------------------------------------------------------
__global__ __launch_bounds__(1024) void topk_xs_kernel(const float* __restrict__ Xt,
                                                       float* __restrict__ y,
                                                       float* __restrict__ Xs, int n) {
    __shared__ float sv[1024];
    __shared__ int   si[1024];
    __shared__ int   s_idx;
    __shared__ float s_val;
    const int tid = threadIdx.x;

    for (int k = 0; k < F; ++k) {
        float bv = -INFINITY;
        int   bi = 0x7fffffff;
        for (int i = tid; i < n; i += 1024) {
            float v = y[i];
            if (v > bv || (v == bv && i < bi)) { bv = v; bi = i; }
        }
        sv[tid] = bv; si[tid] = bi;
        __syncthreads();
        for (int st = 512; st > 0; st >>= 1) {
            if (tid < st) {
                float v2 = sv[tid + st]; int i2 = si[tid + st];
                if (v2 > sv[tid] || (v2 == sv[tid] && i2 < si[tid])) {
                    sv[tid] = v2; si[tid] = i2;
                }
            }
            __syncthreads();
        }
        if (tid == 0) {
            s_idx = si[0];
            s_val = sv[0];
            y[si[0]] = -INFINITY;  // exclude from later rounds
        }
        __syncthreads();
        if (tid < F) Xs[tid * F + k] = Xt[(size_t)s_idx * F + tid] * s_val;
        __syncthreads();
    }
}

// ---------------------------------------------------------------------------
// One 16x16x16 f32 matmul-accumulate = 4 x V_WMMA_F32_16X16X4_F32.
// Layouts (CDNA5 ISA 7.12.2, 32-bit):
//   A 16x4 : lane<16 -> {K=4kc+0,4kc+1}, lane>=16 -> {K=4kc+2,4kc+3}, M=lane%16
//   B 4x16 : VGPR j, lane<16 -> K=4kc+j, lane>=16 -> K=4kc+2+j, N=lane%16
//   C 16x16: VGPR v holds D[v + 8*(lane/16)][lane%16]
// ---------------------------------------------------------------------------
__device__ inline v8f mm16_acc(const float* __restrict__ A, const float* B,
                               v8f c, int half, int l16) {
#pragma unroll
    for (int kc = 0; kc < 4; ++kc) {
        const int k0 = 4 * kc + 2 * half;
        v2f a, b;
        a.x = A[l16 * F + k0];
        a.y = A[l16 * F + k0 + 1];
        b.x = B[k0 * F + l16];
        b.y = B[(k0 + 1) * F + l16];
        c = __builtin_amdgcn_wmma_f32_16x16x4_f32(false, a, false, b,
                                                  (short)0, c, false, false);
    }
    return c;
}

__device__ inline float sigmoid_f(float x) { return 1.f / (1.f + __expf(-x)); }

// ---------------------------------------------------------------------------
// GRU update of W (16x16) with WMMA, then w_eff = W_new @ lin_w. One wave.
// ---------------------------------------------------------------------------
__global__ __launch_bounds__(32) void gru_kernel(
    const float* __restrict__ WZ, const float* __restrict__ UZ, const float* __restrict__ BZ,
    const float* __restrict__ WR, const float* __restrict__ UR, const float* __restrict__ BR,
    const float* __restrict__ WH, const float* __restrict__ UH, const float* __restrict__ BH,
    const float* __restrict__ lin_w,
    float* __restrict__ W_cur, const float* __restrict__ Xs,
    float* __restrict__ w_eff) {
    __shared__ float sRW[F * F];

    const int lane = threadIdx.x;
    const int half = lane >> 4;
    const int l16  = lane & 15;

    // current W in C-fragment layout
    float Wfrag[8];
#pragma unroll
    for (int v = 0; v < 8; ++v) Wfrag[v] = W_cur[(v + 8 * half) * F + l16];

    // Z pre-activation: C := B_Z ; C += W_Z@Xs ; C += U_Z@W
    v8f zc, rc, hc;
#pragma unroll
    for (int v = 0; v < 8; ++v) zc[v] = BZ[(v + 8 * half) * F + l16];
    zc = mm16_acc(WZ, Xs, zc, half, l16);
    zc = mm16_acc(UZ, W_cur, zc, half, l16);

    // R pre-activation
#pragma unroll
    for (int v = 0; v < 8; ++v) rc[v] = BR[(v + 8 * half) * F + l16];
    rc = mm16_acc(WR, Xs, rc, half, l16);
    rc = mm16_acc(UR, W_cur, rc, half, l16);

    // RW = sigmoid(R) * W : C-layout -> LDS (row-major) so it can be a B operand
#pragma unroll
    for (int v = 0; v < 8; ++v)
        sRW[(v + 8 * half) * F + l16] = sigmoid_f(rc[v]) * Wfrag[v];
    __syncthreads();

    // H candidate
#pragma unroll
    for (int v = 0; v < 8; ++v) hc[v] = BH[(v + 8 * half) * F + l16];
    hc = mm16_acc(WH, Xs, hc, half, l16);
    hc = mm16_acc(UH, sRW, hc, half, l16);

    // W_new = (1-Z)*W + Z*tanh(H)
    float wn[8];
#pragma unroll
    for (int v = 0; v < 8; ++v) {
        float z = sigmoid_f(zc[v]);
        wn[v] = (1.f - z) * Wfrag[v] + z * tanhf(hc[v]);
    }
#pragma unroll
    for (int v = 0; v < 8; ++v) W_cur[(v + 8 * half) * F + l16] = wn[v];

    // w_eff[m] = sum_n W_new[m][n] * lin_w[n]  (reduce across 16 lanes/half)
    const float lw = lin_w[l16];
#pragma unroll
    for (int v = 0; v < 8; ++v) {
        float x = wn[v] * lw;
        x += __shfl_xor(x, 1, 32);
        x += __shfl_xor(x, 2, 32);
        x += __shfl_xor(x, 4, 32);
        x += __shfl_xor(x, 8, 32);
        if (l16 == 0) w_eff[v + 8 * half] = x;
    }
}

// ---------------------------------------------------------------------------
// s[i] = X_t[i] . w_eff ;  out_t[i] = lin_b (scatter base)
// ---------------------------------------------------------------------------
__global__ __launch_bounds__(256) void s_kernel(const float* __restrict__ Xt,
                                                const float* __restrict__ w_eff,
                                                const float* __restrict__ lin_b,
                                                float* __restrict__ s,
                                                float* __restrict__ out_t, int n) {
    int i = blockIdx.x * blockDim.x + threadIdx.x;
    if (i >= n) return;
    float acc = 0.f;
#pragma unroll
    for (int f = 0; f < F; ++f) acc += Xt[(size_t)i * F + f] * w_eff[f];
    s[i] = acc;
    out_t[i] = lin_b[0];
}

// ---------------------------------------------------------------------------
// out_t[row[e]] += val[e] * s[col[e]]   (one atomic per edge; L2-resident)
// ---------------------------------------------------------------------------
__global__ __launch_bounds__(256) void scatter_kernel(const int* __restrict__ row,
                                                      const int* __restrict__ col,
                                                      const float* __restrict__ val,
                                                      const float* __restrict__ s,
                                                      float* __restrict__ out_t, int e) {
    int i = blockIdx.x * blockDim.x + threadIdx.x;
    if (i >= e) return;
    atomicAdd(&out_t[row[i]], val[i] * s[col[i]]);
}

// ---------------------------------------------------------------------------
extern "C" void kernel_launch(void* const* d_in, const int* in_sizes, int n_in,
                              void* d_out, int out_size, void* d_ws, size_t ws_size,
                              hipStream_t stream) {
    const float* X      = (const float*)d_in[0];
    const int*   erow   = (const int*)  d_in[1];
    const int*   ecol   = (const int*)  d_in[2];
    const float* eval_  = (const float*)d_in[3];
    const float* p      = (const float*)d_in[4];
    const float* WZ     = (const float*)d_in[5];
    const float* UZ     = (const float*)d_in[6];
    const float* BZ     = (const float*)d_in[7];
    const float* WR     = (const float*)d_in[8];
    const float* UR     = (const float*)d_in[9];
    const float* BR     = (const float*)d_in[10];
    const float* WH     = (const float*)d_in[11];
    const float* UH     = (const float*)d_in[12];
    const float* BH     = (const float*)d_in[13];
    const float* W_init = (const float*)d_in[14];
    const float* lin_w  = (const float*)d_in[15];
    const float* lin_b  = (const float*)d_in[16];
    float* out = (float*)d_out;

    // workspace layout (floats)
    float* ws    = (float*)d_ws;
    float* W_cur = ws;              // 256
    float* Xs    = ws + 256;        // 256
    float* w_eff = ws + 512;        // 16
    float* y     = ws + 1024;       // N
    float* s     = ws + 1024 + NN;  // N

    init_w_kernel<<<1, 256, 0, stream>>>(W_init, W_cur);

    for (int t = 0; t < TT; ++t) {
        const float* Xt = X + (size_t)t * NN * F;
        y_kernel<<<(NN + 255) / 256, 256, 0, stream>>>(Xt, p, y, NN);
        topk_xs_kernel<<<1, 1024, 0, stream>>>(Xt, y, Xs, NN);
        gru_kernel<<<1, 32, 0, stream>>>(WZ, UZ, BZ, WR, UR, BR, WH, UH, BH,
                                         lin_w, W_cur, Xs, w_eff);
        s_kernel<<<(NN + 255) / 256, 256, 0, stream>>>(Xt, w_eff, lin_b, s,
                                                       out + (size_t)t * NN, NN);
        scatter_kernel<<<(EE + 255) / 256, 256, 0, stream>>>(
            erow + (size_t)t * EE, ecol + (size_t)t * EE, eval_ + (size_t)t * EE,
            s, out + (size_t)t * NN, EE);
    }
}